// LengthRegulator_82274393523030
// MI455X (gfx1250) — compile-verified
//
#include <hip/hip_runtime.h>
#include <hip/hip_bf16.h>

// LengthRegulator for MI455X (gfx1250, wave32).
// Pure gather/stream problem: ~201 MB of stores dominate -> optimize for
// b128 vector memory ops with non-temporal stores; scan + index table in LDS.

typedef float v4f __attribute__((ext_vector_type(4)));   // native 128-bit vector

namespace {
constexpr int B       = 32;
constexpr int T       = 512;
constexpr int C       = 384;          // floats per row (1536 bytes)
constexpr int MAX_LEN = 4096;
constexpr int SLICES  = 16;           // output-position slices per batch
constexpr int ROWS    = MAX_LEN / SLICES;   // 256 rows per block
constexpr int NTHREADS = 512;         // 16 wave32s
constexpr int NWAVES   = NTHREADS / 32;
}

__global__ __launch_bounds__(NTHREADS)
void length_regulator_kernel(const float* __restrict__ x,
                             const int*  __restrict__ duration,
                             float* __restrict__ out,
                             void*  __restrict__ mel_ptr,
                             int mel_i64) {
    __shared__ int csum[T];        // inclusive cumsum of durations
    __shared__ int idxs[ROWS];     // searchsorted result per output row

    const int b     = blockIdx.y;
    const int slice = blockIdx.x;
    const int tid   = threadIdx.x;

    // ---- load durations (clamped >= 0) and inclusive scan in LDS ----------
    int d = duration[(size_t)b * T + tid];
    csum[tid] = d > 0 ? d : 0;
    __syncthreads();
    #pragma unroll
    for (int off = 1; off < T; off <<= 1) {
        int v = (tid >= off) ? csum[tid - off] : 0;
        __syncthreads();
        csum[tid] += v;
        __syncthreads();
    }
    const int mel_len = csum[T - 1];

    if (slice == 0 && tid == 0) {
        if (mel_i64) ((long long*)mel_ptr)[b] = (long long)mel_len;
        else         ((float*)mel_ptr)[b]     = (float)mel_len;
    }

    // ---- searchsorted(csum, pos, 'right') for this slice's 256 positions --
    const int pos0 = slice * ROWS;
    if (tid < ROWS) {
        const int pos = pos0 + tid;
        int lo = 0, hi = T;                 // first i with csum[i] > pos
        while (lo < hi) {
            int mid = (lo + hi) >> 1;
            if (csum[mid] <= pos) lo = mid + 1; else hi = mid;
        }
        idxs[tid] = lo < (T - 1) ? lo : (T - 1);   // clip to T-1
    }
    __syncthreads();

    // ---- row copy: one wave per row, 3 x v4f per lane (b128 ops) ----------
    const int wave = tid >> 5;
    const int lane = tid & 31;
    const float* __restrict__ xb = x + (size_t)b * T * C;
    float* __restrict__ ob = out + ((size_t)b * MAX_LEN + pos0) * C;

    for (int r = wave; r < ROWS; r += NWAVES) {
        const int   pos   = pos0 + r;
        const int   idx   = idxs[r];                      // LDS broadcast
        const float scale = (pos < mel_len) ? 1.0f : 0.0f;

        const v4f* __restrict__ src = (const v4f*)(xb + (size_t)idx * C) + lane;
        v4f* __restrict__ dst       = (v4f*)(ob + (size_t)r * C) + lane;

        // Prefetch the row this wave copies on its next iteration (NWAVES rows ahead).
        const int rn = r + NWAVES;
        if (rn < ROWS) {
            const float* pf = xb + (size_t)idxs[rn] * C + lane * 4;
            __builtin_prefetch(pf, 0, 3);                 // global_prefetch_b8
        }

        #pragma unroll
        for (int k = 0; k < 3; ++k) {
            v4f v = src[k * 32];
            v *= scale;
            __builtin_nontemporal_store(v, dst + k * 32); // NT global_store_b128
        }
    }
}

extern "C" void kernel_launch(void* const* d_in, const int* in_sizes, int n_in,
                              void* d_out, int out_size, void* d_ws, size_t ws_size,
                              hipStream_t stream) {
    const float* x        = (const float*)d_in[0];
    const int*   duration = (const int*)d_in[1];   // integer input -> int32 per harness

    constexpr long long MAIN = (long long)B * MAX_LEN * C;  // 50,331,648 floats
    float* out = (float*)d_out;
    void*  mel = (void*)(out + MAIN);
    // Tail holds 32 mel_len values. If the harness left room for 8-byte
    // elements (tail == 64 float-slots), store as int64; else store as float.
    const int mel_i64 = (((long long)out_size - MAIN) >= 64) ? 1 : 0;

    dim3 grid(SLICES, B);
    dim3 block(NTHREADS);
    hipLaunchKernelGGL(length_regulator_kernel, grid, block, 0, stream,
                       x, duration, out, mel, mel_i64);
}